// CRF_28174985462333
// MI455X (gfx1250) — compile-verified
//
#include <hip/hip_runtime.h>
#include <stdint.h>

// CRF Viterbi (max-plus DP), B=32, MAXLEN=256, T=128.
// One workgroup per batch element; thread t owns tag t and holds the full
// transition column trans[:, t] in 128 VGPRs. Feats rows are prefetched one
// step ahead with CDNA5 async global->LDS loads (ASYNCcnt).

namespace {
constexpr int kB      = 32;
constexpr int kMaxLen = 256;
constexpr int kT      = 128;
constexpr int kStart  = kT - 2;   // 126
constexpr int kStop   = kT - 1;   // 127
constexpr float kPen    = -1000.0f;
constexpr float kNegInf = -3.0e38f;
}

// CDNA5 async copy: global -> LDS, 4 bytes per lane, tracked by ASYNCcnt.
__device__ __forceinline__ void async_row_load(const float* gsrc,
                                               const float* lds_dst_generic) {
  // Low 32 bits of a generic LDS pointer are the LDS byte address.
  unsigned lds_off = (unsigned)(uintptr_t)lds_dst_generic;
  asm volatile("global_load_async_to_lds_b32 %0, %1, off"
               :
               : "v"(lds_off), "v"(gsrc)
               : "memory");
}

__device__ __forceinline__ void wait_async0() {
  asm volatile("s_wait_asynccnt 0" ::: "memory");
}

__global__ __launch_bounds__(kT, 1) void crf_viterbi_kernel(
    const float* __restrict__ feats,    // (B, MAXLEN, T)
    const float* __restrict__ trans,    // (T, T)  trans[cur*T + nxt]
    const int*   __restrict__ lengths,  // (B,)
    float*       __restrict__ out)      // (B,)
{
  __shared__ __align__(16) float ap_s[2][kT];   // alpha + feats, ping-pong
  __shared__ __align__(16) float f_s[2][kT];    // async-prefetched feats rows
  __shared__ float wmax_s[kT / 32];

  const int b = blockIdx.x;
  const int t = threadIdx.x;
  const int len = lengths[b];                   // 1..256
  const float* fb = feats + (size_t)b * kMaxLen * kT;

  // Transition column for this thread's nxt tag, kept in VGPRs.
  float tr[kT];
#pragma unroll
  for (int c = 0; c < kT; ++c) tr[c] = trans[c * kT + t];

  // Prefetch feats row 0 (consumed by edge n=1) into f_s[1].
  async_row_load(fb + t, &f_s[1][t]);

  // Edge 0: alpha0 = 0, start penalty on cur != START, no feats term.
  ap_s[0][t] = (t == kStart) ? 0.0f : kPen;
  __syncthreads();

  float alpha;
  {
    float m0 = kNegInf, m1 = kNegInf, m2 = kNegInf, m3 = kNegInf;
    const float4* ap4 = (const float4*)ap_s[0];
#pragma unroll
    for (int c4 = 0; c4 < kT / 4; ++c4) {
      float4 a = ap4[c4];                       // broadcast ds_load_b128
      m0 = fmaxf(m0, a.x + tr[4 * c4 + 0]);
      m1 = fmaxf(m1, a.y + tr[4 * c4 + 1]);
      m2 = fmaxf(m2, a.z + tr[4 * c4 + 2]);
      m3 = fmaxf(m3, a.w + tr[4 * c4 + 3]);
    }
    alpha = fmaxf(fmaxf(m0, m1), fmaxf(m2, m3));
  }

  // Edges n = 1 .. len. Edges past len are identity, so just stop there.
  for (int n = 1; n <= len; ++n) {
    const int buf = n & 1;
    wait_async0();                              // row n-1 landed in f_s[buf]
    if (n < len) {                              // prefetch row n for edge n+1
      async_row_load(fb + (size_t)n * kT + t, &f_s[buf ^ 1][t]);
    }
    ap_s[buf][t] = alpha + f_s[buf][t];         // fold feats[b, n-1, cur]
    __syncthreads();

    float m0 = kNegInf, m1 = kNegInf, m2 = kNegInf, m3 = kNegInf;
    const float4* ap4 = (const float4*)ap_s[buf];
#pragma unroll
    for (int c4 = 0; c4 < kT / 4; ++c4) {
      float4 a = ap4[c4];
      m0 = fmaxf(m0, a.x + tr[4 * c4 + 0]);
      m1 = fmaxf(m1, a.y + tr[4 * c4 + 1]);
      m2 = fmaxf(m2, a.z + tr[4 * c4 + 2]);
      m3 = fmaxf(m3, a.w + tr[4 * c4 + 3]);
    }
    float m = fmaxf(fmaxf(m0, m1), fmaxf(m2, m3));
    if (n == len && t != kStop) m += kPen;      // stop penalty on final edge
    alpha = m;
  }

  // Block-wide max: wave32 shuffle tree, then tiny LDS combine.
#pragma unroll
  for (int off = 16; off >= 1; off >>= 1) {
    alpha = fmaxf(alpha, __shfl_xor(alpha, off, 32));
  }
  if ((t & 31) == 0) wmax_s[t >> 5] = alpha;
  __syncthreads();
  if (t == 0) {
    float r = wmax_s[0];
#pragma unroll
    for (int w = 1; w < kT / 32; ++w) r = fmaxf(r, wmax_s[w]);
    out[b] = r;
  }
}

extern "C" void kernel_launch(void* const* d_in, const int* in_sizes, int n_in,
                              void* d_out, int out_size, void* d_ws, size_t ws_size,
                              hipStream_t stream) {
  const float* feats   = (const float*)d_in[0];  // (32, 256, 128) f32
  const float* trans   = (const float*)d_in[1];  // (128, 128) f32
  const int*   lengths = (const int*)d_in[2];    // (32,) i32
  float* out = (float*)d_out;                    // (32,) f32

  (void)in_sizes; (void)n_in; (void)out_size; (void)d_ws; (void)ws_size;

  crf_viterbi_kernel<<<dim3(kB), dim3(kT), 0, stream>>>(feats, trans, lengths, out);
}